// OpMessagePassing_59768764891819
// MI455X (gfx1250) — compile-verified
//
#include <hip/hip_runtime.h>

#define NN   64
#define DD   32
#define DG   4            // d-channels per workgroup (one float4 slice)
#define ROWP 66           // padded LDS pitch: even (8B-aligned pairs), stride-2 banks
#define PLANE (NN * ROWP) // 4224 floats per channel plane

typedef float v2f __attribute__((ext_vector_type(2)));
typedef float v8f __attribute__((ext_vector_type(8)));

__global__ __launch_bounds__(256)
void mp_bmm_wmma_kernel(const float* __restrict__ A,
                        const float* __restrict__ Bm,
                        const int*   __restrict__ counts,
                        float*       __restrict__ out)
{
    __shared__ float smem[2 * DG * PLANE];   // 132 KB
    float* As = smem;                 // [d][i][k], pitch ROWP
    float* Bt = smem + DG * PLANE;    // [d][j][k], pitch ROWP (B transposed)
    float* Os = smem;                 // reused after compute: [i][j][DG]

    const int b     = blockIdx.y;
    const int dg    = blockIdx.x;     // fastest dim -> 8 d-group blocks of b adjacent (L2 reuse)
    const int dbase = dg * DG;
    const int tid   = threadIdx.x;
    const int cnt   = counts[b];      // uniform

    // ---- Phase 1: stage A (k-masked) and B (transposed) into LDS ----
    const float* Ab = A  + (size_t)b * NN * NN * DD + dbase;
    const float* Bb = Bm + (size_t)b * NN * NN * DD + dbase;

    for (int p = tid; p < NN * NN; p += 256) {
        const int r = p >> 6;         // i for A, k for B
        const int c = p & 63;         // k for A, j for B
        float4 av = *(const float4*)(Ab + (size_t)p * DD);
        const float4 bv = *(const float4*)(Bb + (size_t)p * DD);
        if (c >= cnt) { av.x = 0.f; av.y = 0.f; av.z = 0.f; av.w = 0.f; }
        const int ia  = r * ROWP + c;   // As[i][k]
        const int ibt = c * ROWP + r;   // Bt[j][k]
        As[0 * PLANE + ia]  = av.x;
        As[1 * PLANE + ia]  = av.y;
        As[2 * PLANE + ia]  = av.z;
        As[3 * PLANE + ia]  = av.w;
        Bt[0 * PLANE + ibt] = bv.x;
        Bt[1 * PLANE + ibt] = bv.y;
        Bt[2 * PLANE + ibt] = bv.z;
        Bt[3 * PLANE + ibt] = bv.w;
    }
    __syncthreads();

    // ---- Phase 2: 2x4 register-blocked WMMA per wave ----
    const int lane  = tid & 31;
    const int wave  = tid >> 5;
    const int half  = lane >> 4;      // K-half selector for A/B fragments
    const int l     = lane & 15;
    const int d     = wave >> 1;      // channel 0..3
    const int ihalf = wave & 1;       // row-half: ti in {2*ihalf, 2*ihalf+1}
    const int kSteps = (cnt + 3) >> 2;

    // per-lane base rows (A rows = M = l; Bt rows = N = l)
    const float* Ad = &As[d * PLANE + (ihalf * 32 + l) * ROWP];
    const float* Bd = &Bt[d * PLANE + l * ROWP];

    v8f zero = {0.f, 0.f, 0.f, 0.f, 0.f, 0.f, 0.f, 0.f};
    v8f acc[2][4];
    #pragma unroll
    for (int r = 0; r < 2; ++r)
        #pragma unroll
        for (int c2 = 0; c2 < 4; ++c2) acc[r][c2] = zero;

    for (int ks = 0; ks < kSteps; ++ks) {
        const int k0 = ks * 4 + 2 * half;
        v2f af[2], bf[4];
        #pragma unroll
        for (int r = 0; r < 2; ++r) {           // contiguous pair -> ds_load b64
            af[r].x = Ad[r * 16 * ROWP + k0];
            af[r].y = Ad[r * 16 * ROWP + k0 + 1];
        }
        #pragma unroll
        for (int c2 = 0; c2 < 4; ++c2) {        // contiguous pair -> ds_load b64
            bf[c2].x = Bd[c2 * 16 * ROWP + k0];
            bf[c2].y = Bd[c2 * 16 * ROWP + k0 + 1];
        }
        #pragma unroll
        for (int r = 0; r < 2; ++r)
            #pragma unroll
            for (int c2 = 0; c2 < 4; ++c2)
                acc[r][c2] = __builtin_amdgcn_wmma_f32_16x16x4_f32(
                    false, af[r], false, bf[c2], (short)0, acc[r][c2], false, false);
    }

    __syncthreads();   // all As/Bt reads done; safe to alias Os over As

    // ---- Phase 3: masked scatter of C tiles into Os[i][j][DG] ----
    #pragma unroll
    for (int r = 0; r < 2; ++r) {
        #pragma unroll
        for (int c2 = 0; c2 < 4; ++c2) {
            const int j = c2 * 16 + l;           // C layout: N = lane%16
            #pragma unroll
            for (int v = 0; v < 8; ++v) {
                const int i = ihalf * 32 + r * 16 + v + 8 * half;  // M = v / v+8
                const float val = (i < cnt && j < cnt) ? acc[r][c2][v] : 0.f;
                Os[(i * NN + j) * DG + d] = val;
            }
        }
    }
    __syncthreads();

    // ---- Phase 4: cooperative b128 stores (all DG channels contiguous) ----
    float* ob = out + (size_t)b * NN * NN * DD + dbase;
    for (int p = tid; p < NN * NN; p += 256) {
        *(float4*)(ob + (size_t)p * DD) = *(const float4*)&Os[p * DG];
    }
}

extern "C" void kernel_launch(void* const* d_in, const int* in_sizes, int n_in,
                              void* d_out, int out_size, void* d_ws, size_t ws_size,
                              hipStream_t stream) {
    const float* A      = (const float*)d_in[0];
    const float* B      = (const float*)d_in[1];
    const int*   counts = (const int*)d_in[2];
    float*       out    = (float*)d_out;

    dim3 grid(DD / DG, 256);
    mp_bmm_wmma_kernel<<<grid, 256, 0, stream>>>(A, B, counts, out);
}